// Model_32779190403169
// MI455X (gfx1250) — compile-verified
//
#include <hip/hip_runtime.h>
#include <hip/hip_bf16.h>
#include <stdint.h>

// ---------------------------------------------------------------------------
// Merkle-style tree hash, MI455X (gfx1250).
// Bandwidth-bound: 64 MB int32 leaves streamed once (~2.9us at 23.3 TB/s).
// Data path: CDNA5 async global->LDS movers (global_load_async_to_lds_b128,
// ASYNCcnt / s_wait_asynccnt) with per-instruction-contiguous global spans
// and non-temporal hint; one hash per wave32 lane; in-LDS subtree reduction.
// WMMA is inapplicable: the mix rounds are a serial carry-propagating byte
// recurrence, not a matmul.
// ---------------------------------------------------------------------------

#define T_A          128   // threads per leaf-block  (one leaf PAIR per thread)
#define CHUNK_BYTES  256   // 64 int32 = one leaf pair per thread
#define LDS_STRIDE   272   // 256 + 16 pad: 16B aligned, staggers LDS banks

__device__ __forceinline__ void mix_rounds(uint32_t st[32]) {
#pragma unroll
  for (int r = 0; r < 4; ++r) {
#pragma unroll
    for (int i = 0; i < 32; ++i) {
      st[i] = (st[i] ^ (st[(i + 7) & 31] + st[(i + 13) & 31])) & 255u;
    }
  }
}

__device__ __forceinline__ void pack_words(const uint32_t st[32], uint32_t w[8]) {
#pragma unroll
  for (int q = 0; q < 8; ++q)
    w[q] = st[4 * q] | (st[4 * q + 1] << 8) | (st[4 * q + 2] << 16) |
           (st[4 * q + 3] << 24);
}

// combine two packed 32-byte hashes (8 words each) into one
__device__ __forceinline__ void hash_words(const uint32_t lw[8],
                                           const uint32_t rw[8],
                                           uint32_t ow[8]) {
  uint32_t st[32];
#pragma unroll
  for (int k = 0; k < 32; ++k) {
    uint32_t a = lw[k >> 2] >> ((k & 3) * 8);
    uint32_t b = rw[k >> 2] >> ((k & 3) * 8);
    st[k] = ((a << 5) ^ (b << 5)) & 255u;  // mask kills high-byte garbage
  }
  mix_rounds(st);
  pack_words(st, ow);
}

// ---------------------------------------------------------------------------
// Kernel A: each block consumes 256 leaves (32 KB of int32) and reduces them
// to a single 32-byte hash in LDS.
// ---------------------------------------------------------------------------
__global__ __launch_bounds__(T_A) void k_leaf_blocks(
    const uint32_t* __restrict__ leaves, uint32_t* __restrict__ block_hashes) {
  __shared__ __align__(16) uint8_t inbuf[T_A * LDS_STRIDE];  // 34816 B
  __shared__ uint32_t hw[T_A * 8];                           // 4096 B
  const int tid = threadIdx.x;

  // ---- async global -> LDS copy of this block's 32 KB tile (GVS mode) ----
  // Unit u = 16B granule. Mapping u = it*T_A + tid makes the 32 lanes of each
  // wave-instruction read one contiguous 512B global span (minimal VMEM
  // transactions), while LDS destinations scatter into the padded per-hash
  // chunks: lds = (u/16)*LDS_STRIDE + (u%16)*16.
  const unsigned long long gbase =
      (unsigned long long)(uintptr_t)leaves +
      (unsigned long long)blockIdx.x * (T_A * CHUNK_BYTES);
  // low 32 bits of a flat shared-space address == LDS byte offset
  const uint32_t ldsArr = (uint32_t)(uintptr_t)(&inbuf[0]);
#pragma unroll
  for (int it = 0; it < CHUNK_BYTES / 16; ++it) {
    uint32_t u = (uint32_t)it * T_A + (uint32_t)tid;
    uint32_t g = u * 16u;                                   // contiguous span
    uint32_t l = ldsArr + (u >> 4) * LDS_STRIDE + (u & 15u) * 16u;
    asm volatile("global_load_async_to_lds_b128 %0, %1, %2 th:TH_LOAD_NT"
                 :
                 : "v"(l), "v"(g), "s"(gbase)
                 : "memory");
  }
  asm volatile("s_wait_asynccnt 0x0" ::: "memory");
  __syncthreads();

  // ---- level 0: hash one leaf pair per lane (values are int32 in LDS) ----
  {
    const uint4* cp = (const uint4*)(inbuf + (size_t)tid * LDS_STRIDE);
    uint32_t st[32];
#pragma unroll
    for (int q = 0; q < 8; ++q) {
      uint4 A = cp[q];      // left leaf, values 4q..4q+3
      uint4 B = cp[q + 8];  // right leaf, values 4q..4q+3
      st[4 * q + 0] = ((A.x << 5) ^ (B.x << 5)) & 255u;
      st[4 * q + 1] = ((A.y << 5) ^ (B.y << 5)) & 255u;
      st[4 * q + 2] = ((A.z << 5) ^ (B.z << 5)) & 255u;
      st[4 * q + 3] = ((A.w << 5) ^ (B.w << 5)) & 255u;
    }
    mix_rounds(st);
    uint32_t ow[8];
    pack_words(st, ow);
#pragma unroll
    for (int w = 0; w < 8; ++w) hw[tid * 8 + w] = ow[w];
  }
  __syncthreads();

  // ---- in-block tree reduction: T_A hashes -> 1 ----
  for (int n = T_A / 2; n >= 1; n >>= 1) {
    uint32_t ow[8];
    if (tid < n) {
      uint32_t lw[8], rw[8];
#pragma unroll
      for (int w = 0; w < 8; ++w) {
        lw[w] = hw[(2 * tid) * 8 + w];
        rw[w] = hw[(2 * tid + 1) * 8 + w];
      }
      hash_words(lw, rw, ow);
    }
    __syncthreads();
    if (tid < n) {
#pragma unroll
      for (int w = 0; w < 8; ++w) hw[tid * 8 + w] = ow[w];
    }
    __syncthreads();
  }

  if (tid == 0) {
#pragma unroll
    for (int w = 0; w < 8; ++w) block_hashes[blockIdx.x * 8 + w] = hw[w];
  }
}

// ---------------------------------------------------------------------------
// Kernel B: reduce the per-block hashes (2*n_pairs of them) to the root,
// convert root bytes to float.
// ---------------------------------------------------------------------------
__global__ __launch_bounds__(1024) void k_final(
    const uint32_t* __restrict__ nodes, int n_pairs,
    float* __restrict__ out) {
  __shared__ uint32_t hw[1024 * 8];  // 32 KB, holds first-level outputs
  const int tid = threadIdx.x;

  if (tid < n_pairs) {
    uint32_t lw[8], rw[8], ow[8];
#pragma unroll
    for (int w = 0; w < 8; ++w) {
      lw[w] = nodes[(2 * tid) * 8 + w];
      rw[w] = nodes[(2 * tid + 1) * 8 + w];
    }
    hash_words(lw, rw, ow);
#pragma unroll
    for (int w = 0; w < 8; ++w) hw[tid * 8 + w] = ow[w];
  }
  __syncthreads();

  for (int n = n_pairs >> 1; n >= 1; n >>= 1) {
    uint32_t ow[8];
    if (tid < n) {
      uint32_t lw[8], rw[8];
#pragma unroll
      for (int w = 0; w < 8; ++w) {
        lw[w] = hw[(2 * tid) * 8 + w];
        rw[w] = hw[(2 * tid + 1) * 8 + w];
      }
      hash_words(lw, rw, ow);
    }
    __syncthreads();
    if (tid < n) {
#pragma unroll
      for (int w = 0; w < 8; ++w) hw[tid * 8 + w] = ow[w];
    }
    __syncthreads();
  }

  if (tid == 0) {
#pragma unroll
    for (int k = 0; k < 32; ++k)
      out[k] = (float)((hw[k >> 2] >> ((k & 3) * 8)) & 255u);
  }
}

// ---------------------------------------------------------------------------
extern "C" void kernel_launch(void* const* d_in, const int* in_sizes, int n_in,
                              void* d_out, int out_size, void* d_ws,
                              size_t ws_size, hipStream_t stream) {
  (void)n_in; (void)out_size; (void)ws_size;
  const uint32_t* leaves = (const uint32_t*)d_in[0];  // int32 leaves, [0,256)
  const int n_leaves = in_sizes[0] / 32;              // 524288
  const int blocks = n_leaves / (T_A * 2);            // 2048 (256 leaves/block)
  uint32_t* block_hashes = (uint32_t*)d_ws;           // blocks * 32 bytes

  k_leaf_blocks<<<blocks, T_A, 0, stream>>>(leaves, block_hashes);

  const int n_pairs = blocks / 2;                     // 1024 (<= max block dim)
  k_final<<<1, n_pairs, 0, stream>>>(block_hashes, n_pairs, (float*)d_out);
}